// Net_16286515986870
// MI455X (gfx1250) — compile-verified
//
#include <hip/hip_runtime.h>
#include <hip/hip_bf16.h>
#include <math.h>

typedef float v2f __attribute__((ext_vector_type(2)));
typedef float v8f __attribute__((ext_vector_type(8)));

// ---- order-preserving float<->uint encoding for atomicMax on floats ----
__device__ __forceinline__ unsigned fenc(float f) {
  unsigned u = __float_as_uint(f);
  return (u & 0x80000000u) ? ~u : (u | 0x80000000u);
}
__device__ __forceinline__ float fdec(unsigned e) {
  unsigned u = (e & 0x80000000u) ? (e & 0x7FFFFFFFu) : ~e;
  return __uint_as_float(u);
}

// ---------------- generic zero ----------------
__global__ void k_zero(float* __restrict__ p, long n) {
  long i = (long)blockIdx.x * blockDim.x + threadIdx.x;
  long stride = (long)gridDim.x * blockDim.x;
  for (; i < n; i += stride) p[i] = 0.0f;
}

// ---------------- node encoder: x = nf @ W_ne + b_ne  ([N,9]x[9,16]) ----
__global__ void k_encode_nodes(const float* __restrict__ nf, const float* __restrict__ W,
                               const float* __restrict__ b, float* __restrict__ x, int n) {
  int i = blockIdx.x * blockDim.x + threadIdx.x;
  if (i >= n) return;
  float in[9];
#pragma unroll
  for (int k = 0; k < 9; ++k) in[k] = nf[(size_t)i * 9 + k];
#pragma unroll
  for (int j = 0; j < 16; ++j) {
    float acc = b[j];
#pragma unroll
    for (int k = 0; k < 9; ++k) acc += in[k] * W[k * 16 + j];
    x[(size_t)i * 16 + j] = acc;
  }
}

// -------- edge encoder: ef = ea @ W_ee + b_ee ([E,3]x[3,2]) + column sums ----
__global__ void k_encode_edges(const float* __restrict__ ea, const float* __restrict__ W,
                               const float* __restrict__ b, float* __restrict__ ef,
                               float* __restrict__ ef_sum, int e) {
  __shared__ float s0[256], s1[256];
  float acc0 = 0.f, acc1 = 0.f;
  for (long i = (long)blockIdx.x * blockDim.x + threadIdx.x; i < e;
       i += (long)gridDim.x * blockDim.x) {
    float a0 = ea[i * 3], a1 = ea[i * 3 + 1], a2 = ea[i * 3 + 2];
    float f0 = b[0] + a0 * W[0] + a1 * W[2] + a2 * W[4];
    float f1 = b[1] + a0 * W[1] + a1 * W[3] + a2 * W[5];
    ef[i * 2] = f0; ef[i * 2 + 1] = f1;
    acc0 += f0; acc1 += f1;
  }
  int t = threadIdx.x;
  s0[t] = acc0; s1[t] = acc1;
  __syncthreads();
  for (int off = blockDim.x >> 1; off > 0; off >>= 1) {
    if (t < off) { s0[t] += s0[t + off]; s1[t] += s1[t + off]; }
    __syncthreads();
  }
  if (t == 0) { atomicAdd(&ef_sum[0], s0[0]); atomicAdd(&ef_sum[1], s1[0]); }
}

// -------- per-layer scalars: c_i = We_i^T ae_i (2), loop_alpha_i = mean(ef).c_i ----
__global__ void k_prep(const float* __restrict__ We0, const float* __restrict__ ae0,
                       const float* __restrict__ We1, const float* __restrict__ ae1,
                       const float* __restrict__ We2, const float* __restrict__ ae2,
                       const float* __restrict__ ef_sum, float* __restrict__ scal, int E) {
  if (blockIdx.x != 0 || threadIdx.x != 0) return;
  const float* We[3] = {We0, We1, We2};
  const float* ae[3] = {ae0, ae1, ae2};
  float m0 = ef_sum[0] / (float)E, m1 = ef_sum[1] / (float)E;
  for (int i = 0; i < 3; ++i) {
    float c0 = 0.f, c1 = 0.f;
    for (int k = 0; k < 32; ++k) {
      c0 += We[i][k] * ae[i][k];
      c1 += We[i][32 + k] * ae[i][k];
    }
    scal[i * 2] = c0; scal[i * 2 + 1] = c1;
    scal[6 + i] = m0 * c0 + m1 * c1;  // self-loop a_edge
  }
}

// ---------------- WMMA node GEMM: H[n,32] = A[n,KIN] @ W[KIN,32] (f32) ----------------
template <int KIN>
__global__ void k_gemm_node(const float* __restrict__ A, const float* __restrict__ W,
                            float* __restrict__ H, int n, int nTiles) {
  int wavesPerBlock = blockDim.x >> 5;
  int waveId = blockIdx.x * wavesPerBlock + (threadIdx.x >> 5);
  if (waveId >= nTiles) return;                 // wave-uniform: EXEC stays all-ones
  int lane = threadIdx.x & 31;
  int half = lane >> 4;                          // 0: K 0..1 / M 0..7, 1: K 2..3 / M 8..15
  int mrow = lane & 15;
  int rowBase = waveId * 16;
  int rowA = rowBase + mrow;
  if (rowA >= n) rowA = n - 1;                   // clamp loads for tail tile
  v8f acc0 = {0.f, 0.f, 0.f, 0.f, 0.f, 0.f, 0.f, 0.f};
  v8f acc1 = {0.f, 0.f, 0.f, 0.f, 0.f, 0.f, 0.f, 0.f};
#pragma unroll
  for (int kb = 0; kb < KIN; kb += 4) {
    int k0 = kb + half * 2;
    v2f a, b0, b1;
    a.x  = A[(size_t)rowA * KIN + k0];
    a.y  = A[(size_t)rowA * KIN + k0 + 1];
    b0.x = W[(size_t)k0 * 32 + mrow];
    b0.y = W[(size_t)(k0 + 1) * 32 + mrow];
    b1.x = W[(size_t)k0 * 32 + 16 + mrow];
    b1.y = W[(size_t)(k0 + 1) * 32 + 16 + mrow];
    acc0 = __builtin_amdgcn_wmma_f32_16x16x4_f32(false, a, false, b0, (short)0, acc0, false, false);
    acc1 = __builtin_amdgcn_wmma_f32_16x16x4_f32(false, a, false, b1, (short)0, acc1, false, false);
  }
#pragma unroll
  for (int v = 0; v < 8; ++v) {
    int m = v + half * 8;                        // D layout: VGPR v -> M = v + 8*half
    int orow = rowBase + m;
    if (orow < n) {
      H[(size_t)orow * 32 + mrow]      = acc0[v];
      H[(size_t)orow * 32 + 16 + mrow] = acc1[v];
    }
  }
}

// ---------------- a_src[i] = h[i,:].a_s ; a_dst[i] = h[i,:].a_d ----------------
__global__ void k_node_dots(const float* __restrict__ h, const float* __restrict__ as,
                            const float* __restrict__ ad, float* __restrict__ a_src,
                            float* __restrict__ a_dst, int n) {
  int i = blockIdx.x * blockDim.x + threadIdx.x;
  if (i >= n) return;
  float s = 0.f, d = 0.f;
#pragma unroll
  for (int k = 0; k < 32; ++k) {
    float v = h[(size_t)i * 32 + k];
    s += v * as[k];
    d += v * ad[k];
  }
  a_src[i] = s; a_dst[i] = d;
}

// ---------------- edge pass 1: logits + segment max (atomic on encoded uint) ----
__global__ void k_edge_logits(const int* __restrict__ ei, const float* __restrict__ ef,
                              const float* __restrict__ a_src, const float* __restrict__ a_dst,
                              const float* __restrict__ scal, int layer,
                              float* __restrict__ logits, unsigned* __restrict__ seg_max_u,
                              int E, int Etot) {
  float c0 = scal[layer * 2], c1 = scal[layer * 2 + 1], la = scal[6 + layer];
  for (long e = (long)blockIdx.x * blockDim.x + threadIdx.x; e < Etot;
       e += (long)gridDim.x * blockDim.x) {
    int s, d; float aedge;
    if (e < E) {
      s = ei[e]; d = ei[(long)E + e];
      aedge = ef[e * 2] * c0 + ef[e * 2 + 1] * c1;
    } else {
      s = d = (int)(e - E);
      aedge = la;
    }
    float l = a_src[s] + a_dst[d] + aedge;
    l = (l > 0.f) ? l : 0.2f * l;                // leaky_relu(0.2)
    logits[e] = l;
    atomicMax(&seg_max_u[d], fenc(l));
  }
}

// ---- edge pass 2: wave per edge, lane per channel; unnormalized weighted scatter ----
__global__ void k_edge_scatter(const int* __restrict__ ei, const float* __restrict__ logits,
                               const unsigned* __restrict__ seg_max_u, const float* __restrict__ h,
                               float* __restrict__ seg_sum, float* __restrict__ accum,
                               int E, int Etot) {
  int lane = threadIdx.x & 31;
  long wave = ((long)blockIdx.x * blockDim.x + threadIdx.x) >> 5;
  long nw = ((long)gridDim.x * blockDim.x) >> 5;
  for (long e = wave; e < Etot; e += nw) {
    int s, d;
    if (e < E) { s = ei[e]; d = ei[(long)E + e]; }
    else       { s = d = (int)(e - E); }
    // prefetch next iteration's gather row (gfx1250 global_prefetch_b8)
    long en = e + nw;
    if (en < Etot) {
      int sn = (en < E) ? ei[en] : (int)(en - E);
      __builtin_prefetch(&h[(size_t)sn * 32 + lane], 0, 1);
    }
    float m = fdec(seg_max_u[d]);                // finite: self-loop guarantees >=1 edge
    float ev = __expf(logits[e] - m);            // uniform across the wave
    if (lane == 0) atomicAdd(&seg_sum[d], ev);
    atomicAdd(&accum[(size_t)d * 32 + lane], h[(size_t)s * 32 + lane] * ev);
  }
}

// ---------------- normalize + bias: xout = accum/(sum+eps) + b ----------------
__global__ void k_finalize(float* __restrict__ xout, const float* __restrict__ seg_sum,
                           const float* __restrict__ bb, int n) {
  long total = (long)n * 32;
  for (long idx = (long)blockIdx.x * blockDim.x + threadIdx.x; idx < total;
       idx += (long)gridDim.x * blockDim.x) {
    long node = idx >> 5;
    int k = (int)(idx & 31);
    xout[idx] = xout[idx] / (seg_sum[node] + 1e-16f) + bb[k];
  }
}

// ---------------- mean pool: atomic accumulate concat(x1,x2,x3) per graph ----------------
__global__ void k_pool(const float* __restrict__ x1, const float* __restrict__ x2,
                       const float* __restrict__ x3, const int* __restrict__ batch,
                       float* __restrict__ pool, float* __restrict__ cnt, int n) {
  long total = (long)n * 96;
  for (long idx = (long)blockIdx.x * blockDim.x + threadIdx.x; idx < total;
       idx += (long)gridDim.x * blockDim.x) {
    int node = (int)(idx / 96);
    int c = (int)(idx % 96);
    const float* src = (c < 32) ? x1 : ((c < 64) ? x2 : x3);
    float v = src[(size_t)node * 32 + (c & 31)];
    int g = batch[node];
    atomicAdd(&pool[(size_t)g * 96 + c], v);
    if (c == 0) atomicAdd(&cnt[g], 1.0f);
  }
}

// ---------------- readout: out[g] = (pool[g,:]/max(cnt,1)) @ W_l3 + b_l3 ----------------
__global__ void k_readout(const float* __restrict__ pool, const float* __restrict__ cnt,
                          const float* __restrict__ Wl3, const float* __restrict__ bl3,
                          float* __restrict__ out, int g_count) {
  int g = blockIdx.x * blockDim.x + threadIdx.x;
  if (g >= g_count) return;
  float inv = 1.0f / fmaxf(cnt[g], 1.0f);
  float acc = 0.f;
#pragma unroll 4
  for (int c = 0; c < 96; ++c) acc += pool[(size_t)g * 96 + c] * Wl3[c];
  out[g] = acc * inv + bl3[0];
}

extern "C" void kernel_launch(void* const* d_in, const int* in_sizes, int n_in,
                              void* d_out, int out_size, void* d_ws, size_t ws_size,
                              hipStream_t stream) {
  (void)n_in; (void)ws_size;
  const float* node_feat  = (const float*)d_in[0];
  const int*   edge_index = (const int*)d_in[1];
  const float* edge_attr  = (const float*)d_in[2];
  const int*   batch      = (const int*)d_in[3];
  const float* W_ne = (const float*)d_in[4];
  const float* b_ne = (const float*)d_in[5];
  const float* W_ee = (const float*)d_in[6];
  const float* b_ee = (const float*)d_in[7];
  const float* W_l3 = (const float*)d_in[8];
  const float* b_l3 = (const float*)d_in[9];
  const float *Wl[3], *as_[3], *ad_[3], *We_[3], *ae_[3], *bb_[3];
  for (int i = 0; i < 3; ++i) {
    Wl[i]  = (const float*)d_in[10 + 6 * i];
    as_[i] = (const float*)d_in[11 + 6 * i];
    ad_[i] = (const float*)d_in[12 + 6 * i];
    We_[i] = (const float*)d_in[13 + 6 * i];
    ae_[i] = (const float*)d_in[14 + 6 * i];
    bb_[i] = (const float*)d_in[15 + 6 * i];
  }
  const int N = in_sizes[0] / 9;
  const int E = in_sizes[2] / 3;
  const int G = out_size;
  const int Etot = E + N;

  // ---- workspace layout (floats) ----
  float* p = (float*)d_ws;
  float* x  = p; p += (size_t)N * 16;
  float* h  = p; p += (size_t)N * 32;
  float* x1 = p; p += (size_t)N * 32;
  float* x2 = p; p += (size_t)N * 32;
  float* x3 = p; p += (size_t)N * 32;
  float* a_src = p;   p += N;
  float* a_dst = p;   p += N;
  float* seg_sum = p; p += N;                 // seg_sum + seg_max_u contiguous (zeroed together)
  unsigned* seg_max_u = (unsigned*)p; p += N;
  float* ef = p;     p += (size_t)E * 2;
  float* logits = p; p += Etot;
  float* pool = p;   p += (size_t)G * 96;     // pool + cnt + scal contiguous (zeroed together)
  float* cnt = p;    p += G;
  float* scal = p;   p += 16;                 // [0..5]=c_i, [6..8]=loop_alpha_i, [12..13]=ef_sum
  float* ef_sum = scal + 12;

  // ---- encoders + per-layer scalar precompute ----
  k_zero<<<64, 256, 0, stream>>>(pool, (long)G * 96 + G + 16);
  k_encode_nodes<<<(N + 255) / 256, 256, 0, stream>>>(node_feat, W_ne, b_ne, x, N);
  k_encode_edges<<<512, 256, 0, stream>>>(edge_attr, W_ee, b_ee, ef, ef_sum, E);
  k_prep<<<1, 32, 0, stream>>>(We_[0], ae_[0], We_[1], ae_[1], We_[2], ae_[2], ef_sum, scal, E);

  const float* xin[3] = {x, x1, x2};
  float* xout[3] = {x1, x2, x3};
  int tiles = (N + 15) / 16;
  int gemmBlocks = (tiles + 3) / 4;           // 4 waves (128 threads) per block

  for (int i = 0; i < 3; ++i) {
    k_zero<<<4096, 256, 0, stream>>>(xout[i], (long)N * 32);
    k_zero<<<1024, 256, 0, stream>>>(seg_sum, 2L * N);   // seg_sum + seg_max_u
    if (i == 0)
      k_gemm_node<16><<<gemmBlocks, 128, 0, stream>>>(xin[i], Wl[i], h, N, tiles);
    else
      k_gemm_node<32><<<gemmBlocks, 128, 0, stream>>>(xin[i], Wl[i], h, N, tiles);
    k_node_dots<<<(N + 255) / 256, 256, 0, stream>>>(h, as_[i], ad_[i], a_src, a_dst, N);
    k_edge_logits<<<2048, 256, 0, stream>>>(edge_index, ef, a_src, a_dst, scal, i,
                                            logits, seg_max_u, E, Etot);
    k_edge_scatter<<<2048, 256, 0, stream>>>(edge_index, logits, seg_max_u, h,
                                             seg_sum, xout[i], E, Etot);
    k_finalize<<<4096, 256, 0, stream>>>(xout[i], seg_sum, bb_[i], N);
  }

  k_pool<<<8192, 256, 0, stream>>>(x1, x2, x3, batch, pool, cnt, N);
  k_readout<<<(G + 127) / 128, 128, 0, stream>>>(pool, cnt, W_l3, b_l3, (float*)d_out, G);
}